// TGAP_47674136985681
// MI455X (gfx1250) — compile-verified
//
#include <hip/hip_runtime.h>
#include <math.h>

#define NN 10000   // nodes
#define NE 32000   // edges
#define NB 16      // batch
#define ND 128     // feature dim
#define NH 8       // heads (out)

typedef float v2f __attribute__((ext_vector_type(2)));
typedef float v8f __attribute__((ext_vector_type(8)));

__device__ __forceinline__ float lrelu01(float x) { return x > 0.0f ? x : 0.01f * x; }

// ---------------------------------------------------------------------------
// PGNN layer: elementwise edge scores, segmented softmax over dst, aggregate.
// PASS 1: S1[dst,b,d] += exp(score)      (atomic exp-sum per segment-channel)
// PASS 2: h_n[dst,b,d] += exp(score)/(S1+eps) * g_e   (normalized scatter)
// ---------------------------------------------------------------------------
template <int PASS>
__global__ void pgnn_pass(const float* __restrict__ ent_emb,
                          const float* __restrict__ rel_emb,
                          const float* __restrict__ tau_emb,
                          const float* __restrict__ pgnn_i,
                          const float* __restrict__ pgnn_j,
                          const int* __restrict__ node_idx,
                          const int* __restrict__ edge_src,
                          const int* __restrict__ edge_dst,
                          const int* __restrict__ relation_type,
                          const int* __restrict__ edge_time,
                          const int* __restrict__ batch_time,
                          float* __restrict__ S1,
                          float* __restrict__ hn)
{
  const int e = blockIdx.x;          // one block per edge
  const int d = threadIdx.x;         // 0..127 feature channel
  const int src = edge_src[e];
  const int dst = edge_dst[e];
  int rt = relation_type[e]; rt = rt ? rt : 1;
  int et = edge_time[e];     et = et ? et : 1;
  const float src_d = ent_emb[node_idx[src] * ND + d];
  const float rel_d = rel_emb[rt * ND + d];
  // ai is independent of b: h_n0[n,b,:] == ent_emb[node_idx[n]]
  const float ai = ent_emb[node_idx[dst] * ND + d] * pgnn_i[d];
  const float pj = pgnn_j[d];
  for (int b = 0; b < NB; ++b) {
    int td = et - batch_time[b]; td = td < 0 ? -td : td;
    const float tau_d = tau_emb[(td + 1) * ND + d];
    const float ge    = src_d + rel_d + tau_d;          // g_e[e,b,d]
    const float score = lrelu01(ai * (ge * pj));
    const float p     = expf(score);
    const int   idx   = (dst * NB + b) * ND + d;
    if (PASS == 1) {
      atomicAdd(&S1[idx], p);
    } else {
      const float a = p / (S1[idx] + 1e-16f);
      atomicAdd(&hn[idx], a * ge);
    }
  }
}

__global__ void lrelu_inplace(float* __restrict__ x, int n)
{
  int i = blockIdx.x * blockDim.x + threadIdx.x;
  if (i < n) x[i] = lrelu01(x[i]);
}

// ---------------------------------------------------------------------------
// Dense 16x256 @ 256x128 GEMM via V_WMMA_F32_16X16X4_F32 (full f32 path).
// One wave per 16x16 output tile; 8 waves / block cover all 128 output cols.
// out[m][n] = sum_k X[m][k] * W[n][k] + bias[n]
// A 16x4 f32 layout: lanes m=0..15; v0 holds K=(half?2:0), v1 holds K+1.
// B 4x16  f32 layout: lanes n=0..15; v0 holds K=(half?2:0), v1 holds K+1.
// C/D 16x16 f32: vgpr v -> row (v + half*8), col = lane&15.
// ---------------------------------------------------------------------------
__device__ __forceinline__ void gemm_16x256_wmma(float (*X)[256],
                                                 const float* __restrict__ W,
                                                 const float* __restrict__ bias,
                                                 float* __restrict__ out)
{
  const int lane = threadIdx.x & 31;
  const int tile = threadIdx.x >> 5;       // 0..7
  const int n0   = tile * 16;
  const int half = lane >> 4;              // 0 or 1
  const int l    = lane & 15;
  v8f c = {};
  for (int k0 = 0; k0 < 256; k0 += 4) {
    const int k = k0 + half * 2;
    v2f a, b;
    a.x = X[l][k];
    a.y = X[l][k + 1];
    b.x = W[(n0 + l) * 256 + k];
    b.y = W[(n0 + l) * 256 + k + 1];
    c = __builtin_amdgcn_wmma_f32_16x16x4_f32(false, a, false, b, (short)0, c,
                                              false, false);
  }
  const int col = n0 + l;
  const float bv = bias[col];
#pragma unroll
  for (int v = 0; v < 8; ++v) {
    const int row = v + half * 8;
    out[row * ND + col] = c[v] + bv;
  }
}

__global__ void wmma_query(const float* __restrict__ ent_emb,
                           const float* __restrict__ rel_emb,
                           const float* __restrict__ Wc_w,
                           const float* __restrict__ Wc_b,
                           const int* __restrict__ head,
                           const int* __restrict__ relation,
                           float* __restrict__ query)
{
  __shared__ float X[16][256];
  for (int i = threadIdx.x; i < 16 * 256; i += 256) {
    const int b = i >> 8, k = i & 255;
    X[b][k] = (k < ND) ? ent_emb[head[b] * ND + k]
                       : rel_emb[relation[b] * ND + (k - ND)];
  }
  __syncthreads();
  gemm_16x256_wmma(X, Wc_w, Wc_b, query);
}

__global__ void wmma_ghead(const float* __restrict__ hn,
                           const float* __restrict__ query,
                           const float* __restrict__ Wn_w,
                           const float* __restrict__ Wn_b,
                           const int* __restrict__ head,
                           float* __restrict__ g_head)
{
  __shared__ float X[16][256];
  for (int i = threadIdx.x; i < 16 * 256; i += 256) {
    const int b = i >> 8, k = i & 255;
    X[b][k] = (k < ND) ? hn[(head[b] * NB + b) * ND + k]
                       : query[b * ND + (k - ND)];
  }
  __syncthreads();
  gemm_16x256_wmma(X, Wn_w, Wn_b, g_head);
}

// ---------------------------------------------------------------------------
// Attention-flow phase. g_n / a are nonzero only at (head[b], b), so only
// edges with src == head[b] contribute (~E*B/N ~= 51 total).
// PASS 1: S2[b,h] += exp(score2)   PASS 2: a_new[dst,b,h] += exp/(S2+eps)
// ---------------------------------------------------------------------------
template <int PASS>
__global__ void attn_pass(const float* __restrict__ rel_emb,
                          const float* __restrict__ tau_emb,
                          const float* __restrict__ g_head,
                          const float* __restrict__ hn,
                          const float* __restrict__ attn_i,
                          const float* __restrict__ attn_j,
                          const float* __restrict__ inattn_i,
                          const float* __restrict__ inattn_j,
                          const int* __restrict__ edge_src,
                          const int* __restrict__ edge_dst,
                          const int* __restrict__ relation_type,
                          const int* __restrict__ edge_time,
                          const int* __restrict__ head,
                          const int* __restrict__ batch_time,
                          float* __restrict__ S2,
                          float* __restrict__ a_new)
{
  const int i = blockIdx.x * blockDim.x + threadIdx.x;
  if (i >= NE * NB) return;
  const int e = i >> 4;
  const int b = i & 15;
  const int src = edge_src[e];
  if (src != head[b]) return;

  const int dst = edge_dst[e];
  int rt = relation_type[e]; rt = rt ? rt : 1;
  int et = edge_time[e];     et = et ? et : 1;
  int td = et - batch_time[b]; td = td < 0 ? -td : td;
  const float* rel  = rel_emb + rt * ND;
  const float* tau  = tau_emb + (td + 1) * ND;
  const float* gh   = g_head + b * ND;
  const float* hdst = hn + (dst * NB + b) * ND;
  const bool dst_is_head = (dst == head[b]);

  for (int h = 0; h < NH; ++h) {
    float ta = 0.0f, ti = 0.0f;
#pragma unroll
    for (int dd = 0; dd < 16; ++dd) {
      const int d = h * 16 + dd;
      const float base = rel[d] + tau[d];
      const float gsub = base + (dst_is_head ? gh[d] : 0.0f);  // g_e_sub
      const float gout = base + hdst[d];                        // g_e_out
      const float ghd  = gh[d];                                 // g_n at src
      ta += (ghd * attn_i[d])   * (gsub * attn_j[d]);
      ti += (ghd * inattn_i[d]) * (gout * inattn_j[d]);
    }
    const float s = lrelu01(ta) + lrelu01(ti);
    const float p = expf(s);
    if (PASS == 1) {
      atomicAdd(&S2[b * NH + h], p);
    } else {
      const float t = p / (S2[b * NH + h] + 1e-16f);
      atomicAdd(&a_new[(dst * NB + b) * NH + h], t);
    }
  }
}

// ---------------------------------------------------------------------------
extern "C" void kernel_launch(void* const* d_in, const int* in_sizes, int n_in,
                              void* d_out, int out_size, void* d_ws, size_t ws_size,
                              hipStream_t stream)
{
  const float* ent_emb  = (const float*)d_in[0];
  const float* rel_emb  = (const float*)d_in[1];
  const float* tau_emb  = (const float*)d_in[2];
  const float* Wc_w     = (const float*)d_in[3];
  const float* Wc_b     = (const float*)d_in[4];
  const float* Wn_w     = (const float*)d_in[5];
  const float* Wn_b     = (const float*)d_in[6];
  const float* attn_i   = (const float*)d_in[7];
  const float* attn_j   = (const float*)d_in[8];
  const float* inattn_i = (const float*)d_in[9];
  const float* inattn_j = (const float*)d_in[10];
  const float* pgnn_i   = (const float*)d_in[11];
  const float* pgnn_j   = (const float*)d_in[12];
  const int* node_idx      = (const int*)d_in[13];
  const int* edge_src      = (const int*)d_in[14];
  const int* edge_dst      = (const int*)d_in[15];
  const int* relation_type = (const int*)d_in[16];
  const int* edge_time     = (const int*)d_in[17];
  const int* head          = (const int*)d_in[18];
  const int* relation      = (const int*)d_in[19];
  const int* batch_time    = (const int*)d_in[20];

  float* h_n   = (float*)d_out;                       // [N,B,D]
  float* a_new = h_n + (size_t)NN * NB * ND;          // [N,B,H]

  float* S1     = (float*)d_ws;                       // [N,B,D] exp-sums
  float* query  = S1 + (size_t)NN * NB * ND;          // [16,128]
  float* g_head = query + NB * ND;                    // [16,128]
  float* S2     = g_head + NB * ND;                   // [16,8]

  const size_t ws_used = ((size_t)NN * NB * ND + 2 * NB * ND + NB * NH) * sizeof(float);
  hipMemsetAsync(d_ws, 0, ws_used, stream);
  hipMemsetAsync(d_out, 0, (size_t)out_size * sizeof(float), stream);

  // query does not depend on h_n; issue early.
  wmma_query<<<1, 256, 0, stream>>>(ent_emb, rel_emb, Wc_w, Wc_b, head,
                                    relation, query);

  pgnn_pass<1><<<NE, ND, 0, stream>>>(ent_emb, rel_emb, tau_emb, pgnn_i, pgnn_j,
                                      node_idx, edge_src, edge_dst, relation_type,
                                      edge_time, batch_time, S1, h_n);
  pgnn_pass<2><<<NE, ND, 0, stream>>>(ent_emb, rel_emb, tau_emb, pgnn_i, pgnn_j,
                                      node_idx, edge_src, edge_dst, relation_type,
                                      edge_time, batch_time, S1, h_n);
  const int nhd = NN * NB * ND;
  lrelu_inplace<<<(nhd + 255) / 256, 256, 0, stream>>>(h_n, nhd);

  wmma_ghead<<<1, 256, 0, stream>>>(h_n, query, Wn_w, Wn_b, head, g_head);

  const int neb = NE * NB;
  attn_pass<1><<<(neb + 255) / 256, 256, 0, stream>>>(
      rel_emb, tau_emb, g_head, h_n, attn_i, attn_j, inattn_i, inattn_j,
      edge_src, edge_dst, relation_type, edge_time, head, batch_time, S2, a_new);
  attn_pass<2><<<(neb + 255) / 256, 256, 0, stream>>>(
      rel_emb, tau_emb, g_head, h_n, attn_i, attn_j, inattn_i, inattn_j,
      edge_src, edge_dst, relation_type, edge_time, head, batch_time, S2, a_new);
}